// ConnectionistTemporalClassification_90039694394099
// MI455X (gfx1250) — compile-verified
//
#include <hip/hip_runtime.h>

// Problem constants (fixed by the reference file)
#define TMAXC 2048
#define VOCAB 41
#define LLAB  256
#define SDIM  514    // S+1 alpha cells, indices 0..513 (S = 2*256+1 = 513)
#define PADA  2      // halo cells so a[s-1], a[s-2] reads need no branches
#define BLOCK 544    // 17 wave32 waves; threads 0..513 own one real state each

#if defined(__HIP_DEVICE_COMPILE__) && __has_builtin(__builtin_amdgcn_exp2f)
#define EXP2F(x) __builtin_amdgcn_exp2f(x)   // v_exp_f32
#else
#define EXP2F(x) exp2f(x)                    // HIP host+device overload
#endif
#if defined(__HIP_DEVICE_COMPILE__) && __has_builtin(__builtin_amdgcn_logf)
#define LOG2F(x) __builtin_amdgcn_logf(x)    // v_log_f32 (log2)
#else
#define LOG2F(x) log2f(x)                    // HIP host+device overload
#endif

#define LOG2E 1.4426950408889634f
#define LN2   0.6931471805599453f

// logaddexp: max + log1p(exp(-|x-y|)); computed with v_exp_f32/v_log_f32.
// 1+t in [1,2] so plain log of (1+t) loses <= ~1 ulp absolute vs log1p.
__device__ __forceinline__ float lax(float x, float y) {
    float mx = fmaxf(x, y);
    float t  = EXP2F(fabsf(x - y) * -LOG2E);
    return __builtin_fmaf(LOG2F(1.0f + t), LN2, mx);
}

__global__ __launch_bounds__(BLOCK)
void ctc_alpha_kernel(const float* __restrict__ lp,     // [B, TMAX, V] log-probs
                      const int*   __restrict__ phone,  // [B, L]
                      const int*   __restrict__ length, // [B]
                      float*       __restrict__ lls)    // [B] out
{
    __shared__ float A[2][BLOCK + PADA];  // double-buffered alpha rows (+2 halo)
    __shared__ int   ph[LLAB];

    const int b = blockIdx.x;
    const int s = threadIdx.x;
    const float* __restrict__ lpb = lp + (size_t)b * (TMAXC * VOCAB);
    const int T = length[b];

    // CDNA5 async global->LDS copy of the phone labels (no VGPR round-trip).
    // addrspace(3) offset == low 32 bits of the flat shared-aperture address.
    if (s < LLAB) {
        unsigned lds_off = (unsigned)(uintptr_t)(&ph[s]);
        const int* gsrc  = phone + b * LLAB + s;
        asm volatile("global_load_async_to_lds_b32 %0, %1, off"
                     :: "v"(lds_off), "v"(gsrc) : "memory");
    }
    if (s < PADA) { A[0][s] = 0.0f; A[1][s] = 0.0f; }  // halo: never observable
    asm volatile("s_wait_asynccnt 0x0" ::: "memory");
    __syncthreads();

    // Per-state constants (reference semantics incl. clip()); threads >= SDIM
    // compute harmless pad values so the hot loop is branchless.
    const int  half = s >> 1;
    const bool odd  = (s & 1) != 0;
    int i1 = half - 1; i1 = i1 < 0 ? 0 : (i1 > LLAB - 1 ? LLAB - 1 : i1);
    int i2 = half - 2; i2 = i2 < 0 ? 0 : (i2 > LLAB - 1 ? LLAB - 1 : i2);
    const bool two_term = odd || ((half > 1) && (ph[i1] == ph[i2]));
    const int  id = odd ? 0 : (ph[i1] + 1);

    // a1: zeros everywhere (faithful quirk), except [1]=log_y[0,0], [2]=log_y[0,ph[0]+1]
    {
        float v = 0.0f;
        if (s == 1) v = lpb[0];
        if (s == 2) v = lpb[ph[0] + 1];
        A[0][PADA + s] = v;
    }
    __syncthreads();

    const int steps = T - 1;            // scan consumes ys rows 1..T-1
    float y = lpb[VOCAB + id];          // gathered log_y[1, ids[s]]

    // One fused step: read src row, write dst row, advance gathered y; one barrier.
    auto step = [&](const float* __restrict__ src, float* __restrict__ dst, int t) {
        const int tn = (t + 1 < steps) ? (t + 1) : steps;
        const float ynext = lpb[tn * VOCAB + id];   // register prefetch across barrier
        if (s < VOCAB) {
            const int tp = (tn + 1 < steps) ? (tn + 1) : steps;
            __builtin_prefetch(lpb + tp * VOCAB + s, 0, 0);  // global_prefetch_b8
        }
        const float a   = src[PADA + s];
        const float am1 = src[PADA + s - 1];
        const float am2 = src[PADA + s - 2];
        const float two   = lax(a, am1);
        const float three = lax(two, am2);
        float nv = (two_term ? two : three) + y;
        nv = (s == 0) ? 0.0f : nv;      // column 0 pinned to log-prob 1
        dst[PADA + s] = nv;
        y = ynext;
        __syncthreads();
    };

    // Unroll 2 steps/iteration: immediate LDS offsets, no buffer-index math.
    int t = 1;
    for (; t + 1 <= steps; t += 2) {
        step(A[0], A[1], t);
        step(A[1], A[0], t + 1);
    }
    const bool odd_tail = (t <= steps); // uniform per block (T is per-batch)
    if (odd_tail) step(A[0], A[1], t);
    const float* fin = odd_tail ? A[1] : A[0];

    // fin holds all_rows[T-1]; ll = logaddexp(aT[S], aT[S-1])
    if (s == 0) {
        lls[b] = lax(fin[PADA + SDIM - 1], fin[PADA + SDIM - 2]);
    }
}

// Deterministic fixed-order reduction: out = -sum(lls)
__global__ void ctc_reduce_kernel(const float* __restrict__ lls,
                                  float* __restrict__ out, int n)
{
    __shared__ float buf[256];
    const int i = threadIdx.x;
    float acc = 0.0f;
    for (int j = i; j < n; j += 256) acc += lls[j];
    buf[i] = acc;
    __syncthreads();
    for (int off = 128; off > 0; off >>= 1) {
        if (i < off) buf[i] += buf[i + off];
        __syncthreads();
    }
    if (i == 0) out[0] = -buf[0];
}

extern "C" void kernel_launch(void* const* d_in, const int* in_sizes, int n_in,
                              void* d_out, int out_size, void* d_ws, size_t ws_size,
                              hipStream_t stream) {
    const float* lp     = (const float*)d_in[0];  // [B, TMAX, V] fp32
    const int*   phone  = (const int*)  d_in[1];  // [B, L] int32
    const int*   length = (const int*)  d_in[2];  // [B] int32
    const int B = in_sizes[2];

    float* lls = (float*)d_ws;                    // B floats of scratch

    ctc_alpha_kernel<<<B, BLOCK, 0, stream>>>(lp, phone, length, lls);
    ctc_reduce_kernel<<<1, 256, 0, stream>>>(lls, (float*)d_out, B);
}